// MultiQueryAttention_15693810499842
// MI455X (gfx1250) — compile-verified
//
#include <hip/hip_runtime.h>
#include <hip/hip_bf16.h>

// out[b,s,h,:] == (x @ Wv + bv)[b,s,:]  for every head h  (see analysis:
// MQA broadcast before RoPE/attention makes softmax uniform -> out = v tiled 12x).
// So this is a single [M=65536, K=768] x [K=768, N=64] fp32 GEMM + bias,
// with the 64-wide result broadcast 12x along the last axis.

#define K_DIM   768
#define N_DIM   64
#define NHEADS  12
#define M_TILE  128
#define KC      64
#define SA_LD   68   // padded LDS row stride (floats) for A tile / C staging
#define SBT_LD  68   // padded LDS row stride (floats) for B^T chunk

typedef float v2f __attribute__((ext_vector_type(2)));
typedef float v4f __attribute__((ext_vector_type(4)));
typedef float v8f __attribute__((ext_vector_type(8)));

__launch_bounds__(256, 2)
__global__ void mqa_vproj_wmma_kernel(const float* __restrict__ x,
                                      const float* __restrict__ Wv,
                                      const float* __restrict__ bv,
                                      float* __restrict__ out) {
  __shared__ float sA [M_TILE * SA_LD];   // 34816 B : A tile, later C staging
  __shared__ float sBT[N_DIM  * SBT_LD];  // 17408 B : Wv chunk, transposed [n][k]

  const int tid  = threadIdx.x;
  const int lane = tid & 31;
  const int wave = tid >> 5;
  const int mBlock = blockIdx.x * M_TILE;

  const int l15 = lane & 15;   // row/col within fragment
  const int hi  = lane >> 4;   // 0: K+0..1, 1: K+2..3 (ISA 16x4 A / 4x16 B layout)

  // cooperative staging coords (one 16-lane group per row, float4 per lane)
  const int rA = tid >> 4;     // 0..15
  const int cA = tid & 15;     // col group, col = cA*4

  // bias for the epilogue's float4 column group (bv has 16 float4 groups)
  const v4f bb = *(const v4f*)(bv + cA * 4);

  v8f acc0 = {}, acc1 = {}, acc2 = {}, acc3 = {};

  for (int kc = 0; kc < K_DIM; kc += KC) {
    // ---- stage A tile: x[mBlock..+127][kc..+63] -> sA (coalesced b128 loads)
    #pragma unroll
    for (int p = 0; p < 8; ++p) {
      const int row = p * 16 + rA;
      v4f xv = *(const v4f*)(x + (size_t)(mBlock + row) * K_DIM + kc + cA * 4);
      *(v4f*)(sA + row * SA_LD + cA * 4) = xv;
    }
    // ---- stage Wv chunk transposed: Wv[kc+kk][n] -> sBT[n][kk]
    {
      const int n  = tid & 63;   // coalesced along n in global memory
      const int k4 = tid >> 6;   // 0..3
      #pragma unroll
      for (int p = 0; p < 16; ++p) {
        const int kk = p * 4 + k4;
        sBT[n * SBT_LD + kk] = Wv[(size_t)(kc + kk) * N_DIM + n];
      }
    }
    __syncthreads();

    const int mBase = wave * 16;
    const float* aPtr = sA  + (mBase + l15) * SA_LD  + hi * 2;
    const float* bPtr = sBT + l15 * SBT_LD + hi * 2;

    #pragma unroll
    for (int kk = 0; kk < KC; kk += 4) {
      v2f aF = *(const v2f*)(aPtr + kk);
      v2f b0 = *(const v2f*)(bPtr + 0 * 16 * SBT_LD + kk);
      v2f b1 = *(const v2f*)(bPtr + 1 * 16 * SBT_LD + kk);
      v2f b2 = *(const v2f*)(bPtr + 2 * 16 * SBT_LD + kk);
      v2f b3 = *(const v2f*)(bPtr + 3 * 16 * SBT_LD + kk);
      // D = A(16x4 f32) * B(4x16 f32) + C   -> v_wmma_f32_16x16x4_f32
      acc0 = __builtin_amdgcn_wmma_f32_16x16x4_f32(false, aF, false, b0, (short)0, acc0, false, false);
      acc1 = __builtin_amdgcn_wmma_f32_16x16x4_f32(false, aF, false, b1, (short)0, acc1, false, false);
      acc2 = __builtin_amdgcn_wmma_f32_16x16x4_f32(false, aF, false, b2, (short)0, acc2, false, false);
      acc3 = __builtin_amdgcn_wmma_f32_16x16x4_f32(false, aF, false, b3, (short)0, acc3, false, false);
    }
    __syncthreads();
  }

  // ---- stage C through LDS (reuse sA) for coalesced b128 output stores.
  // C/D layout: VGPR r, lanes 0-15 -> M=r, N=lane; lanes 16-31 -> M=8+r, N=lane-16.
  {
    const int mBase = wave * 16;
    #pragma unroll
    for (int r = 0; r < 8; ++r) {
      const int row = mBase + r + hi * 8;
      sA[row * SA_LD + 0 * 16 + l15] = acc0[r];
      sA[row * SA_LD + 1 * 16 + l15] = acc1[r];
      sA[row * SA_LD + 2 * 16 + l15] = acc2[r];
      sA[row * SA_LD + 3 * 16 + l15] = acc3[r];
    }
  }
  __syncthreads();

  // ---- out[m][h*64 + n] = C[m][n] + bv[n], broadcast over h = 0..11
  #pragma unroll
  for (int p = 0; p < 8; ++p) {
    const int row = p * 16 + rA;
    v4f c = *(const v4f*)(sA + row * SA_LD + cA * 4);
    c += bb;
    float* orow = out + (size_t)(mBlock + row) * (NHEADS * N_DIM) + cA * 4;
    #pragma unroll
    for (int h = 0; h < NHEADS; ++h) {
      *(v4f*)(orow + h * N_DIM) = c;
    }
  }
}

extern "C" void kernel_launch(void* const* d_in, const int* in_sizes, int n_in,
                              void* d_out, int out_size, void* d_ws, size_t ws_size,
                              hipStream_t stream) {
  // setup_inputs order: x, Wq, bq, Wk, bk, Wv, bv  (all float32)
  const float* x  = (const float*)d_in[0];
  const float* Wv = (const float*)d_in[5];
  const float* bv = (const float*)d_in[6];
  float* out = (float*)d_out;

  const int M = in_sizes[0] / K_DIM;           // B*S = 65536
  const int grid = M / M_TILE;                 // 512 blocks of 256 threads (8 waves)
  mqa_vproj_wmma_kernel<<<grid, 256, 0, stream>>>(x, Wv, bv, out);
}